// HakesPQ_75247827025932
// MI455X (gfx1250) — compile-verified
//
#include <hip/hip_runtime.h>
#include <hip/hip_bf16.h>

// PQ nearest-codeword forward:
//   k* = argmax_k ( <v, c_k> - 0.5*||c_k||^2 ),  out[n, m*16+d] = cb[m][k*][d]
// N=32768, D=768, M=48 subspaces, DSUB=16, KSUB=16.
// Roofline: 200 MB traffic @ 23.3 TB/s ~ 8.6 us; compute trivial -> bandwidth bound.
// WMMA role swap: A = codebook (codes x K), B = vec tile (K x vectors), so the
// argmax over codes is VGPR-local (branchless cndmask) + one xor-16 shuffle.

#define PQ_M    48
#define PQ_KSUB 16
#define PQ_DSUB 16
#define PQ_D    (PQ_M * PQ_DSUB)      // 768
#define ROWPAD  20                    // LDS row stride (floats): 16B-aligned rows, 16 distinct banks

typedef __attribute__((ext_vector_type(2))) float v2f;
typedef __attribute__((ext_vector_type(4))) float v4f;
typedef __attribute__((ext_vector_type(8))) float v8f;

__global__ __launch_bounds__(256) void
hakes_pq_wmma_kernel(const float* __restrict__ vecs,
                     const float* __restrict__ codebooks,
                     float* __restrict__ out)
{
    __shared__ float lds_cb[PQ_M * PQ_KSUB * ROWPAD];   // 61440 B, padded codebook
    __shared__ float lds_c2[PQ_M * PQ_KSUB];            //  3072 B, -0.5*||c_k||^2

    // ---- Stage codebook + fused bias into LDS (uniform: 768 = 3*256) ----
    for (int r = threadIdx.x; r < PQ_M * PQ_KSUB; r += 256) {
        const float* src = codebooks + (size_t)r * PQ_DSUB;   // 64B-aligned
        float sum = 0.0f;
        #pragma unroll
        for (int d = 0; d < PQ_DSUB; ++d) {
            float x = src[d];
            lds_cb[r * ROWPAD + d] = x;
            sum += x * x;
        }
        lds_c2[r] = -0.5f * sum;      // bias folded into the accumulator init
    }
    __syncthreads();

    const int wave = threadIdx.x >> 5;
    const int lane = threadIdx.x & 31;
    const int half = lane >> 4;       // 0: lanes 0-15 (codes 0-7), 1: lanes 16-31 (codes 8-15)
    const int lh   = lane & 15;       // vector column within the 16-wide tile

    const int n0 = blockIdx.x * 128 + wave * 16;
    const float* __restrict__ vrow = vecs + (size_t)(n0 + lh) * PQ_D + 2 * half;

    for (int m = 0; m < PQ_M; ++m) {
        // ---- acc init: C[v][lane] = -0.5*||c_{v+8*half}||^2 (two ds_load_b128) ----
        const float* __restrict__ c2p = &lds_c2[m * PQ_KSUB + 8 * half];
        v8f acc;
        #pragma unroll
        for (int v = 0; v < 8; ++v) acc[v] = c2p[v];

        // ---- score = cb(16x16) * vecsT(16x16) via 4 chained K=4 f32 WMMAs ----
        // A (codes x K): lane lh holds code row lh; lanes0-15 K={4j,4j+1}, lanes16-31 K={4j+2,4j+3}
        const float* __restrict__ ap = &lds_cb[(m * PQ_KSUB + lh) * ROWPAD + 2 * half];
        // B (K x vectors): lane lh holds vector column n0+lh, same K split
        const float* __restrict__ bp = vrow + m * PQ_DSUB;
        #pragma unroll
        for (int j = 0; j < 4; ++j) {
            v2f a = *(const v2f*)(ap + 4 * j);   // ds_load_b64, 8B-aligned
            v2f b = *(const v2f*)(bp + 4 * j);   // global_load_b64
            acc = __builtin_amdgcn_wmma_f32_16x16x4_f32(
                false, a, false, b, (short)0, acc, false, false);
        }

        // ---- branchless argmax over codes: VGPR-local (k = v + 8*half) ----
        float best = acc[0];
        int   bk   = 0;
        #pragma unroll
        for (int v = 1; v < 8; ++v) {
            bool t = acc[v] > best;              // strict '>' keeps lowest index on ties
            best = t ? acc[v] : best;
            bk   = t ? v      : bk;
        }
        int kmine = bk + 8 * half;

        // merge the two halves (codes 0-7 vs 8-15): one xor-16 exchange
        float ob  = __shfl_xor(best,  16, 32);
        int   ok  = __shfl_xor(kmine, 16, 32);
        bool take = half ? (ob >= best) : (ob > best);   // lower half owns lower k on ties
        int kstar = take ? ok : kmine;                   // same value in lane pair

        // ---- emit winning codeword: lane writes its half-row (32B) ----
        const float* __restrict__ cw = &lds_cb[(m * PQ_KSUB + kstar) * ROWPAD + 8 * half];
        v4f w0 = *(const v4f*)(cw);          // ds_load_b128 (16B-aligned: ROWPAD=20)
        v4f w1 = *(const v4f*)(cw + 4);
        float* __restrict__ op = out + (size_t)(n0 + lh) * PQ_D + m * PQ_DSUB + 8 * half;
        *(v4f*)(op)     = w0;                // global_store_b128
        *(v4f*)(op + 4) = w1;
    }
}

extern "C" void kernel_launch(void* const* d_in, const int* in_sizes, int n_in,
                              void* d_out, int out_size, void* d_ws, size_t ws_size,
                              hipStream_t stream) {
    (void)n_in; (void)out_size; (void)d_ws; (void)ws_size;
    const float* vecs      = (const float*)d_in[0];   // (N, 768) f32
    const float* codebooks = (const float*)d_in[1];   // (48, 16, 16) f32
    float*       out       = (float*)d_out;           // (N, 768) f32

    const int n_vecs = in_sizes[0] / PQ_D;            // 32768
    const int blocks = n_vecs / 128;                  // 128 vectors per block (8 waves x 16)
    hakes_pq_wmma_kernel<<<blocks, 256, 0, stream>>>(vecs, codebooks, out);
}